// SpatialGraphConv_52175262711998
// MI455X (gfx1250) — compile-verified
//
#include <hip/hip_runtime.h>
#include <hip/hip_bf16.h>
#include <math.h>

// ---------------------------------------------------------------------------
// Sizes (fixed by the reference): B=8, N=96, D=16, H0=H1=50, OUT=64
//   W1: (51,50)  rows 0:16 Wxi | 16:32 Wxj | 32:48 Wxk | 48 w_ij | 49 w_jk | 50 w_ik
//   W2: (83,50)  rows 0:16 W2a | 16:32 W2b | 32 W2c | 33:83 W2d
//   W3: (66,64)
// ---------------------------------------------------------------------------

typedef float v2f __attribute__((ext_vector_type(2)));
typedef float v8f __attribute__((ext_vector_type(8)));

static __device__ __forceinline__ v8f wmma4(v2f a, v2f b, v8f c) {
  // V_WMMA_F32_16X16X4_F32: D(16x16 f32) = A(16x4 f32) * B(4x16 f32) + C
  return __builtin_amdgcn_wmma_f32_16x16x4_f32(
      /*neg_a=*/false, a, /*neg_b=*/false, b,
      /*c_mod=*/(short)0, c, /*reuse_a=*/false, /*reuse_b=*/false);
}

static __device__ __forceinline__ float leaky(float v) {
  return v >= 0.f ? v : 0.05f * v;
}

// Workspace layout (float offsets)
#define WS_DIS   0        // B*96*96 = 73728
#define WS_A     73728    // B*96*50 = 38400   x @ Wxi
#define WS_BM    112128   //                    x @ Wxj
#define WS_C     150528   //                    x @ Wxk
#define WS_P     188928   //                    x @ W2a
#define WS_Q     227328   //                    x @ W2b
#define WS_SUMC  265728   //                    adj @ C
#define WS_SUMQ  304128   //                    adj @ Q
#define WS_DEG   342528   // B*96 = 768
#define WS_SDAD  343296   // B*96   sum_j adj*dis
// total 344064 floats = 1,376,256 bytes

// ---------------------------------------------------------------------------
// K1: pairwise distances + per-node reductions deg, sum(adj*dis)
// grid = B*96 (b,i), block = 96 (j)
// ---------------------------------------------------------------------------
__global__ void k_pre1(const float* __restrict__ x, const float* __restrict__ adj,
                       float* __restrict__ dis, float* __restrict__ deg,
                       float* __restrict__ sdad) {
  const int bi = blockIdx.x;          // b*96 + i
  const int b  = bi / 96;
  const int j  = threadIdx.x;
  const float* xi = x + bi * 16;
  const float* xj = x + (b * 96 + j) * 16;
  float s = 0.f;
#pragma unroll
  for (int d = 0; d < 16; ++d) { float df = xi[d] - xj[d]; s += df * df; }
  const float dv = sqrtf(s + 1e-10f);
  dis[(size_t)bi * 96 + j] = dv;
  const float av = adj[(size_t)bi * 96 + j];

  __shared__ float r1[96], r2[96];
  r1[j] = av; r2[j] = av * dv;
  __syncthreads();
  if (j == 0) {
    float s1 = 0.f, s2 = 0.f;
    for (int k = 0; k < 96; ++k) { s1 += r1[k]; s2 += r2[k]; }
    deg[bi] = s1; sdad[bi] = s2;
  }
}

// ---------------------------------------------------------------------------
// K2: per-node projections A,Bm,C (from W1) and P,Q (from W2), all (B*96,50)
// grid = B*96, block = 64 (g < 50 active)
// ---------------------------------------------------------------------------
__global__ void k_pre2(const float* __restrict__ x, const float* __restrict__ W1,
                       const float* __restrict__ W2,
                       float* __restrict__ Aa, float* __restrict__ Bm,
                       float* __restrict__ Cc, float* __restrict__ Pp,
                       float* __restrict__ Qq) {
  const int bn = blockIdx.x;
  const int g  = threadIdx.x;
  if (g >= 50) return;
  const float* xr = x + bn * 16;
  float a = 0.f, bm = 0.f, c = 0.f, p = 0.f, q = 0.f;
#pragma unroll
  for (int d = 0; d < 16; ++d) {
    const float xv = xr[d];
    a  += xv * W1[d * 50 + g];
    bm += xv * W1[(16 + d) * 50 + g];
    c  += xv * W1[(32 + d) * 50 + g];
    p  += xv * W2[d * 50 + g];
    q  += xv * W2[(16 + d) * 50 + g];
  }
  Aa[bn * 50 + g] = a;  Bm[bn * 50 + g] = bm;  Cc[bn * 50 + g] = c;
  Pp[bn * 50 + g] = p;  Qq[bn * 50 + g] = q;
}

// ---------------------------------------------------------------------------
// K3: sumC = adj @ C, sumQ = adj @ Q  (per batch (96,96)@(96,50))
// grid = B*96, block = 64 (g < 50 active)
// ---------------------------------------------------------------------------
__global__ void k_pre3(const float* __restrict__ adj, const float* __restrict__ Cc,
                       const float* __restrict__ Qq,
                       float* __restrict__ sumC, float* __restrict__ sumQ) {
  const int bi = blockIdx.x;
  const int b  = bi / 96;
  const int g  = threadIdx.x;
  if (g >= 50) return;
  const float* ar = adj + (size_t)bi * 96;
  float sc = 0.f, sq = 0.f;
  for (int k = 0; k < 96; ++k) {
    const float av = ar[k];
    sc += av * Cc[(b * 96 + k) * 50 + g];
    sq += av * Qq[(b * 96 + k) * 50 + g];
  }
  sumC[bi * 50 + g] = sc;
  sumQ[bi * 50 + g] = sq;
}

// ---------------------------------------------------------------------------
// Main fused kernel: grid = B*6 (batch, i-tile of 16), block = 192 (6 waves)
//  Phase 1: sd_ik tile (16x96) via WMMA f32 16x16x4 (wave w -> j-tile w)
//  Phase 2: M[i,g] = sum_j adj*leaky(adj*S)   (elementwise, per-wave partials)
//  Phase 3: T = M @ W2d via WMMA (waves 0-3 -> n-tiles)
//  Phase 4: out = leaky([x | leaky(m2_sum)] @ W3 + b3) via WMMA
// ---------------------------------------------------------------------------
__launch_bounds__(192)
__global__ void k_main(const float* __restrict__ x, const float* __restrict__ adj,
                       const float* __restrict__ W1, const float* __restrict__ b1,
                       const float* __restrict__ W2, const float* __restrict__ b2,
                       const float* __restrict__ W3, const float* __restrict__ b3,
                       const float* __restrict__ ws, float* __restrict__ out) {
  const int b  = blockIdx.x / 6;
  const int ib = (blockIdx.x % 6) * 16;
  const int t  = threadIdx.x;
  const int wv = t >> 5;          // wave id 0..5
  const int l  = t & 31;          // lane id
  const int li = l & 15;
  const int lh = l >> 4;          // half-wave 0/1
  const int kh = 2 * lh;          // K offset within WMMA A/B fragment

  const float* dis  = ws + WS_DIS;
  const float* Aa   = ws + WS_A;
  const float* Bm   = ws + WS_BM;
  const float* Pp   = ws + WS_P;
  const float* sumC = ws + WS_SUMC;
  const float* sumQ = ws + WS_SUMQ;
  const float* deg  = ws + WS_DEG;
  const float* sdad = ws + WS_SDAD;

  __shared__ float s_sdik[16 * 96];
  __shared__ float s_A[16 * 50];
  __shared__ float s_deg[96], s_sdad[96];
  __shared__ float s_b1[50], s_wij[50], s_wjk[50], s_wik[50];
  __shared__ float s_Mpart[6 * 16 * 50];
  __shared__ float s_Mpad[16 * 52];   // K padded 50 -> 52
  __shared__ float s_T[16 * 64];
  __shared__ float s_fin[16 * 68];    // [x(16) | m2_sum(50) | pad(2)]

  // ---- Phase 0: stage per-block constants into LDS ----
  for (int idx = t; idx < 16 * 50; idx += 192) {
    const int i = idx / 50, g = idx % 50;
    s_A[idx] = Aa[(b * 96 + ib + i) * 50 + g];
  }
  for (int idx = t; idx < 96; idx += 192) {
    s_deg[idx]  = deg[b * 96 + idx];
    s_sdad[idx] = sdad[b * 96 + idx];
  }
  for (int idx = t; idx < 50; idx += 192) {
    s_b1[idx]  = b1[idx];
    s_wij[idx] = W1[48 * 50 + idx];
    s_wjk[idx] = W1[49 * 50 + idx];
    s_wik[idx] = W1[50 * 50 + idx];
  }

  // ---- Phase 1: sd_ik[i, j] = sum_k dis[i,k] * adj[j,k]  (WMMA, K=96) ----
  {
    const int jb = wv * 16;
    v8f acc = {};
    const float* dA = dis + (size_t)(b * 96 + ib + li) * 96;  // A row (M = li)
    const float* dB = adj + (size_t)(b * 96 + jb + li) * 96;  // B col (N = li) = adj row
#pragma unroll
    for (int kb = 0; kb < 96; kb += 4) {
      v2f a, bb;
      a.x  = dA[kb + kh];     a.y  = dA[kb + kh + 1];
      bb.x = dB[kb + kh];     bb.y = dB[kb + kh + 1];
      acc = wmma4(a, bb, acc);
    }
#pragma unroll
    for (int r = 0; r < 8; ++r) {
      const int m = r + 8 * lh;             // C/D row
      s_sdik[m * 96 + jb + li] = acc[r];    // C/D col = li
    }
  }
  __syncthreads();

  // ---- Phase 2: per-wave partial M over its 16 j's ----
  {
    const int jb = wv * 16;
    const int i  = li;            // local row
    const int g0 = lh * 25;       // half of H0 per half-wave
    float Macc[25];
#pragma unroll
    for (int gg = 0; gg < 25; ++gg) Macc[gg] = 0.f;

    for (int jj = 0; jj < 16; ++jj) {
      const int j = jb + jj;
      const float aij = adj[(size_t)(b * 96 + ib + i) * 96 + j];
      const float dij = dis[(size_t)(b * 96 + ib + i) * 96 + j];
      const float sik = s_sdik[i * 96 + j];
      const float dj  = s_deg[j];
      const float sj  = s_sdad[j];
      const float* BmJ = Bm   + (b * 96 + j) * 50;
      const float* sCJ = sumC + (b * 96 + j) * 50;
#pragma unroll
      for (int gg = 0; gg < 25; ++gg) {
        const int g = g0 + gg;
        const float S = dj * (s_A[i * 50 + g] + BmJ[g] + s_b1[g] + dij * s_wij[g])
                      + sCJ[g] + sj * s_wjk[g] + sik * s_wik[g];
        const float m3 = leaky(aij * S);   // m3_sum element
        Macc[gg] += aij * m3;              // adj-weighted row-sum over j
      }
    }
#pragma unroll
    for (int gg = 0; gg < 25; ++gg)
      s_Mpart[(wv * 16 + i) * 50 + g0 + gg] = Macc[gg];
  }
  __syncthreads();

  // ---- Fixed-order reduction of wave partials -> Mpad (zero-padded K) ----
  for (int idx = t; idx < 16 * 52; idx += 192) {
    const int i = idx / 52, g = idx % 52;
    float s = 0.f;
    if (g < 50) {
#pragma unroll
      for (int w = 0; w < 6; ++w) s += s_Mpart[(w * 16 + i) * 50 + g];
    }
    s_Mpad[idx] = s;
  }
  __syncthreads();

  // ---- Phase 3: T = Mpad(16x52) @ W2d(50x50, zero-padded)  (WMMA) ----
  if (wv < 4) {
    const int nb = wv * 16;
    const int n  = nb + li;
    v8f acc = {};
#pragma unroll
    for (int kb = 0; kb < 52; kb += 4) {
      v2f a, bb;
      a.x = s_Mpad[li * 52 + kb + kh];
      a.y = s_Mpad[li * 52 + kb + kh + 1];
      const int k0 = kb + kh, k1 = kb + kh + 1;
      bb.x = (k0 < 50 && n < 50) ? W2[(33 + k0) * 50 + n] : 0.f;
      bb.y = (k1 < 50 && n < 50) ? W2[(33 + k1) * 50 + n] : 0.f;
      acc = wmma4(a, bb, acc);
    }
#pragma unroll
    for (int r = 0; r < 8; ++r) {
      const int m = r + 8 * lh;
      s_T[m * 64 + nb + li] = acc[r];
    }
  }
  __syncthreads();

  // ---- m2_sum (closed form) + build fin = [x_i | leaky(m2_sum) | 0 0] ----
  for (int idx = t; idx < 16 * 68; idx += 192) {
    const int i = idx / 68, c = idx % 68;
    float v;
    if (c < 16) {
      v = x[(b * 96 + ib + i) * 16 + c];
    } else if (c < 66) {
      const int h = c - 16;
      const float di = s_deg[ib + i];
      const float m2s = di * (Pp[(b * 96 + ib + i) * 50 + h] + b2[h])
                      + sumQ[(b * 96 + ib + i) * 50 + h]
                      + s_sdad[ib + i] * W2[32 * 50 + h]
                      + s_T[i * 64 + h];
      v = leaky(m2s);
    } else {
      v = 0.f;
    }
    s_fin[idx] = v;
  }
  __syncthreads();

  // ---- Phase 4: out = leaky(fin(16x66) @ W3(66x64) + b3)  (WMMA) ----
  if (wv < 4) {
    const int nb = wv * 16;
    const int n  = nb + li;
    const float bias = b3[n];
    v8f acc;
#pragma unroll
    for (int r = 0; r < 8; ++r) acc[r] = bias;   // C element (m,n) = b3[n]
#pragma unroll
    for (int kb = 0; kb < 68; kb += 4) {
      v2f a, bb;
      a.x = s_fin[li * 68 + kb + kh];
      a.y = s_fin[li * 68 + kb + kh + 1];
      const int k0 = kb + kh, k1 = kb + kh + 1;
      bb.x = (k0 < 66) ? W3[k0 * 64 + n] : 0.f;
      bb.y = (k1 < 66) ? W3[k1 * 64 + n] : 0.f;
      acc = wmma4(a, bb, acc);
    }
#pragma unroll
    for (int r = 0; r < 8; ++r) {
      const int m = r + 8 * lh;
      out[(b * 96 + ib + m) * 64 + nb + li] = leaky(acc[r]);
    }
  }
}

// ---------------------------------------------------------------------------
extern "C" void kernel_launch(void* const* d_in, const int* in_sizes, int n_in,
                              void* d_out, int out_size, void* d_ws, size_t ws_size,
                              hipStream_t stream) {
  const float* x   = (const float*)d_in[0];
  const float* adj = (const float*)d_in[1];
  const float* W1  = (const float*)d_in[2];
  const float* b1  = (const float*)d_in[3];
  const float* W2  = (const float*)d_in[4];
  const float* b2  = (const float*)d_in[5];
  const float* W3  = (const float*)d_in[6];
  const float* b3  = (const float*)d_in[7];
  float* ws  = (float*)d_ws;
  float* out = (float*)d_out;

  float* dis  = ws + WS_DIS;
  float* Aa   = ws + WS_A;
  float* Bm   = ws + WS_BM;
  float* Cc   = ws + WS_C;
  float* Pp   = ws + WS_P;
  float* Qq   = ws + WS_Q;
  float* sumC = ws + WS_SUMC;
  float* sumQ = ws + WS_SUMQ;
  float* deg  = ws + WS_DEG;
  float* sdad = ws + WS_SDAD;

  k_pre1<<<768, 96, 0, stream>>>(x, adj, dis, deg, sdad);
  k_pre2<<<768, 64, 0, stream>>>(x, W1, W2, Aa, Bm, Cc, Pp, Qq);
  k_pre3<<<768, 64, 0, stream>>>(adj, Cc, Qq, sumC, sumQ);
  k_main<<<48, 192, 0, stream>>>(x, adj, W1, b1, W2, b2, W3, b3, ws, out);
}